// DynaConv_34720515620992
// MI455X (gfx1250) — compile-verified
//
#include <hip/hip_runtime.h>
#include <math.h>

typedef float v2f __attribute__((ext_vector_type(2)));
typedef float v4f __attribute__((ext_vector_type(4)));
typedef float v8f __attribute__((ext_vector_type(8)));

#define NB 4            // batch
#define NC 16           // in channels
#define NO 32           // out channels
#define HW 256
#define LPIX (HW*HW)    // 65536 pixels per image
#define DF 144          // C*9
#define DQ 288          // O2*9 (expanded K for GEMM2)
#define WAVES 4
#define TPW 8           // tiles per wave
#define ROWS_TOTAL (NB*LPIX)            // 262144
#define TILES_TOTAL (ROWS_TOTAL/16)     // 16384
#define BLOCKS (TILES_TOTAL/(WAVES*TPW))// 512

static __device__ __forceinline__ v8f wmma4(v2f a, v2f b, v8f c) {
  // D = A(16x4 f32) * B(4x16 f32) + C(16x16 f32)
  return __builtin_amdgcn_wmma_f32_16x16x4_f32(
      false, a, false, b, (short)0, c, false, false);
}

__global__ __launch_bounds__(WAVES * 32)
void dynaconv_kernel(const float* __restrict__ X,
                     const float* __restrict__ W1,
                     const float* __restrict__ b1,
                     const float* __restrict__ W2,
                     const float* __restrict__ b2,
                     const float* __restrict__ bias,
                     float* __restrict__ Out) {
  // Shared weight stages (interleaved pair layout: [(k>>1)*64 + n*2 + (k&1)])
  __shared__ __align__(16) float sW1[ (DF/2) * 64 ];   // 4608  f32
  __shared__ __align__(16) float sW2[ (DQ/2) * 64 ];   // 9216  f32
  // Per-wave scratch
  __shared__ __align__(16) float sFeat[WAVES][16 * DF];   // 16x144
  __shared__ __align__(16) float sPc  [WAVES][16 * 9];
  __shared__ __align__(16) float sHid [WAVES][16 * NO];   // also reused as out-transpose

  const int tid  = threadIdx.x;
  const int lane = tid & 31;
  const int wv   = tid >> 5;
  const int col  = lane & 15;   // N index within tile / A-row served
  const int h    = lane >> 4;   // half-wave selector (K sub-pair)

  // ---- stage W1: sW1[(f>>1)*64 + o*2 + (f&1)] = W1[o,f], coalesced reads
  for (int e = tid; e < NO * DF; e += WAVES * 32) {
    int o = e / DF, f = e - o * DF;
    sW1[(f >> 1) * 64 + (o << 1) + (f & 1)] = W1[e];
  }
  // ---- stage W2': q = o2*9+j ; sW2[(q>>1)*64 + o*2 + (q&1)] = W2[(o*9+j)*32 + o2]
  for (int e = tid; e < DQ * NO; e += WAVES * 32) {
    int g = e >> 5, o2 = e & 31;
    int o = g / 9, j = g - o * 9;
    int q = o2 * 9 + j;
    sW2[(q >> 1) * 64 + (o << 1) + (q & 1)] = W2[e];
  }
  // per-lane constant preloads
  const float b1v0 = b1[col],      b1v1 = b1[col + 16];
  const float bs0  = bias[col],    bs1  = bias[col + 16];
  float b2a[9], b2b[9];
#pragma unroll
  for (int j = 0; j < 9; ++j) { b2a[j] = b2[col * 9 + j]; b2b[j] = b2[(col + 16) * 9 + j]; }
  __syncthreads();

  float* feat = sFeat[wv];
  float* pc   = sPc[wv];
  float* hid  = sHid[wv];

  for (int it = 0; it < TPW; ++it) {
    const int t    = ((int)blockIdx.x * WAVES + wv) * TPW + it;
    const int row0 = t << 4;
    const int bimg = row0 >> 16;       // row0 / 65536
    const int p0   = row0 & (LPIX - 1);

    // ---- phase A: im2col into LDS (zero-padded), feat[m][c*9+ij]
    for (int e = lane; e < 16 * DF; e += 32) {
      int m = e / DF, f = e - m * DF;
      int c = f / 9, ij = f - c * 9;
      int di = ij / 3, dj = ij - di * 3;
      int p = p0 + m;
      int yy = (p >> 8) + di - 1;
      int xx = (p & 255) + dj - 1;
      float v = 0.0f;
      if ((unsigned)yy < (unsigned)HW && (unsigned)xx < (unsigned)HW)
        v = X[(((bimg * NC + c) * HW + yy) * HW) + xx];
      feat[e] = v;
    }
    __syncthreads();

    // ---- phase B: pc[m][j] = sum_c feat[m][c*9+j]
    for (int e = lane; e < 16 * 9; e += 32) {
      int m = e / 9, j = e - m * 9;
      float s = 0.0f;
#pragma unroll
      for (int c = 0; c < NC; ++c) s += feat[m * DF + c * 9 + j];
      pc[e] = s;
    }
    __syncthreads();

    // ---- phase C: GEMM1  hid = tanh(feat @ W1^T + b1)
    v8f acc0, acc1;
#pragma unroll
    for (int r = 0; r < 8; ++r) { acc0[r] = b1v0; acc1[r] = b1v1; }
    const float* aBase  = feat + col * DF + 2 * h;      // A(m=col, k=4s+2h..)
    const float* w1Base = sW1 + h * 64 + (col << 1);
#pragma unroll
    for (int s = 0; s < DF / 4; ++s) {
      v2f a  = *(const v2f*)(aBase + 4 * s);
      v2f bA = *(const v2f*)(w1Base + s * 128);
      v2f bB = *(const v2f*)(w1Base + s * 128 + 32);
      acc0 = wmma4(a, bA, acc0);
      acc1 = wmma4(a, bB, acc1);
    }
#pragma unroll
    for (int r = 0; r < 8; ++r) {
      int m2 = r + 8 * h;
      hid[m2 * NO + col]      = tanhf(acc0[r]);
      hid[m2 * NO + col + 16] = tanhf(acc1[r]);
    }
    __syncthreads();

    // ---- phase D: pull row (m=col) of hid and pc into registers
    float hr[NO];
#pragma unroll
    for (int k = 0; k < NO / 4; ++k) {
      v4f v = *(const v4f*)(hid + col * NO + 4 * k);
      hr[4 * k + 0] = v.x; hr[4 * k + 1] = v.y; hr[4 * k + 2] = v.z; hr[4 * k + 3] = v.w;
    }
    float pr[9];
#pragma unroll
    for (int j = 0; j < 9; ++j) pr[j] = pc[col * 9 + j];

    // C init for GEMM2: bias[o] + sum_j pc[m][j]*b2[o*9+j]
#pragma unroll
    for (int r = 0; r < 8; ++r) {
      int m2 = r + 8 * h;
      float s0 = bs0, s1 = bs1;
#pragma unroll
      for (int j = 0; j < 9; ++j) { float pv = pc[m2 * 9 + j]; s0 += pv * b2a[j]; s1 += pv * b2b[j]; }
      acc0[r] = s0; acc1[r] = s1;
    }

    // ---- phase E: GEMM2  out = z @ W2' ;  z(q)=hr[q/9]*pr[q%9] built in-registers
    const float* w2Base = sW2 + h * 64 + (col << 1);
#pragma unroll
    for (int s = 0; s < DQ / 4; ++s) {
      const int q = 4 * s;
      float z0 = hr[(q + 0) / 9] * pr[(q + 0) % 9];
      float z1 = hr[(q + 1) / 9] * pr[(q + 1) % 9];
      float z2 = hr[(q + 2) / 9] * pr[(q + 2) % 9];
      float z3 = hr[(q + 3) / 9] * pr[(q + 3) % 9];
      v2f a; a.x = h ? z2 : z0; a.y = h ? z3 : z1;
      v2f bA = *(const v2f*)(w2Base + s * 128);
      v2f bB = *(const v2f*)(w2Base + s * 128 + 32);
      acc0 = wmma4(a, bA, acc0);
      acc1 = wmma4(a, bB, acc1);
    }

    // ---- phase F: transpose through LDS, coalesced store
    float* outT = hid;  // reuse region (16x32)
#pragma unroll
    for (int r = 0; r < 8; ++r) {
      int m2 = r + 8 * h;
      outT[m2 * NO + col]      = acc0[r];
      outT[m2 * NO + col + 16] = acc1[r];
    }
    __syncthreads();
#pragma unroll
    for (int e = 0; e < 16; ++e) {
      int idx = e * 32 + lane;          // 512 = 16m x 32o
      int o = idx >> 4, m = idx & 15;
      Out[((long)(bimg * NO + o)) * LPIX + p0 + m] = outT[m * NO + o];
    }
    __syncthreads();
  }
}

extern "C" void kernel_launch(void* const* d_in, const int* in_sizes, int n_in,
                              void* d_out, int out_size, void* d_ws, size_t ws_size,
                              hipStream_t stream) {
  (void)in_sizes; (void)n_in; (void)out_size; (void)d_ws; (void)ws_size;
  const float* X    = (const float*)d_in[0];
  const float* W1   = (const float*)d_in[1];
  const float* b1   = (const float*)d_in[2];
  const float* W2   = (const float*)d_in[3];
  const float* b2   = (const float*)d_in[4];
  const float* bias = (const float*)d_in[5];
  float* Out = (float*)d_out;
  dynaconv_kernel<<<BLOCKS, WAVES * 32, 0, stream>>>(X, W1, b1, W2, b2, bias, Out);
}